// GCN_27986006901444
// MI455X (gfx1250) — compile-verified
//
#include <hip/hip_runtime.h>

#define N_NODES 100000
#define N_EDGES 1600000
#define D 64

typedef float v2f __attribute__((ext_vector_type(2)));
typedef float v8f __attribute__((ext_vector_type(8)));

// ---------------------------------------------------------------------------
// Phase 1: symmetric-normalization degree.  deg[i] = 1 (self loop) + #edges
// with col == i, then in-place deg[i] := rsqrt(deg[i]).
// ---------------------------------------------------------------------------
__global__ void GCN_deg_init(float* __restrict__ deg) {
    int i = blockIdx.x * blockDim.x + threadIdx.x;
    if (i < N_NODES) deg[i] = 1.0f;  // self loop contributes 1
}

__global__ void GCN_deg_count(const int* __restrict__ col, float* __restrict__ deg) {
    int e = blockIdx.x * blockDim.x + threadIdx.x;
    if (e < N_EDGES) {
        __hip_atomic_fetch_add(&deg[col[e]], 1.0f,
                               __ATOMIC_RELAXED, __HIP_MEMORY_SCOPE_AGENT);
    }
}

__global__ void GCN_deg_to_dinv(float* __restrict__ deg) {
    int i = blockIdx.x * blockDim.x + threadIdx.x;
    if (i < N_NODES) deg[i] = rsqrtf(deg[i]);  // deg >= 1 always
}

// ---------------------------------------------------------------------------
// Phase 2: h = x @ W via V_WMMA_F32_16X16X4_F32.
// Block = 256 threads = 8 waves; each wave owns a 16x64 row-slab of h
// (block covers 128 rows).  W (64x64 f32, 16 KB) staged in LDS.
//
// Each wave keeps FOUR 16x16 f32 accumulators (columns n0 = 0,16,32,48) so
// one A fragment load feeds 4 WMMAs.  64 WMMAs per wave covers 16x64.
//
// Fragment layouts (ISA 7.12.2, 32-bit):
//   A 16x4 : lane l -> M = l&15 ; VGPR0/1 = K={0,1} (l<16) or K={2,3} (l>=16)
//   B 4x16 : lane l -> N = l&15 ; VGPR0/1 = K={0,1} (l<16) or K={2,3} (l>=16)
//   C 16x16: VGPR j -> M = j + 8*(l>=16), N = l&15
// ---------------------------------------------------------------------------
__global__ __launch_bounds__(256)
void GCN_gemm_wmma(const float* __restrict__ x, const float* __restrict__ W,
                   float* __restrict__ h) {
    __shared__ float Ws[64 * 64];

    const int t = threadIdx.x;
    // Stage W into LDS: 4096 floats / 256 threads = 4 x float4 per thread.
    const float4* Wv  = (const float4*)W;
    float4*       Wsv = (float4*)Ws;
#pragma unroll
    for (int i = 0; i < 4; ++i) Wsv[t + i * 256] = Wv[t + i * 256];
    __syncthreads();

    const int wave = t >> 5;
    const int lane = t & 31;
    const int half = lane >> 4;   // 0: K-pair {0,1}, 1: K-pair {2,3}
    const int l16  = lane & 15;

    const long m0 = (long)blockIdx.x * 128 + wave * 16;

    // A-row for this lane; clamp out-of-range rows (EXEC must stay all-1s
    // across WMMA), their results are discarded at store time.
    long rowA = m0 + l16;
    if (rowA >= N_NODES) rowA = N_NODES - 1;
    const float* __restrict__ xr = x + (size_t)rowA * D;

    v8f c0 = {0.f, 0.f, 0.f, 0.f, 0.f, 0.f, 0.f, 0.f};
    v8f c1 = c0, c2 = c0, c3 = c0;

#pragma unroll
    for (int k = 0; k < 64; k += 4) {
        const int kk = k + half * 2;
        v2f a = *(const v2f*)(xr + kk);          // global_load_b64

        v2f b0, b1, b2, b3;
        b0.x = Ws[kk * 64 + l16];                // ds loads (2addr-pairable)
        b0.y = Ws[(kk + 1) * 64 + l16];
        b1.x = Ws[kk * 64 + 16 + l16];
        b1.y = Ws[(kk + 1) * 64 + 16 + l16];
        b2.x = Ws[kk * 64 + 32 + l16];
        b2.y = Ws[(kk + 1) * 64 + 32 + l16];
        b3.x = Ws[kk * 64 + 48 + l16];
        b3.y = Ws[(kk + 1) * 64 + 48 + l16];

        // (neg_a, A, neg_b, B, c_mod, C, reuse_a, reuse_b)
        c0 = __builtin_amdgcn_wmma_f32_16x16x4_f32(false, a, false, b0, (short)0, c0, false, false);
        c1 = __builtin_amdgcn_wmma_f32_16x16x4_f32(false, a, false, b1, (short)0, c1, false, false);
        c2 = __builtin_amdgcn_wmma_f32_16x16x4_f32(false, a, false, b2, (short)0, c2, false, false);
        c3 = __builtin_amdgcn_wmma_f32_16x16x4_f32(false, a, false, b3, (short)0, c3, false, false);
    }

    // Write back D: VGPR j -> row m0 + 8*half + j; cols l16 + {0,16,32,48}.
    const long rbase = m0 + half * 8;
#pragma unroll
    for (int j = 0; j < 8; ++j) {
        long r = rbase + j;
        if (r < N_NODES) {
            float* __restrict__ hr = h + r * D;
            hr[l16]      = c0[j];
            hr[l16 + 16] = c1[j];
            hr[l16 + 32] = c2[j];
            hr[l16 + 48] = c3[j];
        }
    }
}

// ---------------------------------------------------------------------------
// Phase 3a: out[i,f] = b[f] + dinv[i]^2 * h[i,f]   (bias + self-loop message)
// ---------------------------------------------------------------------------
__global__ void GCN_out_init(const float* __restrict__ h, const float* __restrict__ dinv,
                             const float* __restrict__ b, float* __restrict__ out) {
    long i = (long)blockIdx.x * blockDim.x + threadIdx.x;
    if (i < (long)N_NODES * D) {
        long node = i >> 6;
        int  f    = (int)(i & 63);
        float di  = dinv[node];
        out[i] = b[f] + h[i] * (di * di);
    }
}

// ---------------------------------------------------------------------------
// Phase 3b: edge scatter.  64 consecutive lanes per edge -> coalesced 256 B
// gather of h[row] and coalesced fp32 atomic-adds into out[col].  h and out
// (~52 MB) are L2-resident on MI455X (192 MB L2), so both the gather and the
// atomic RMW resolve in L2.
// ---------------------------------------------------------------------------
__global__ __launch_bounds__(256)
void GCN_scatter(const int* __restrict__ ei, const float* __restrict__ h,
                 const float* __restrict__ dinv, float* __restrict__ out) {
    const int  t = threadIdx.x;
    const long e = (long)blockIdx.x * 4 + (t >> 6);
    const int  f = t & 63;
    if (e < N_EDGES) {
        const int r    = ei[e];             // source node
        const int cdst = ei[N_EDGES + e];   // target node
        const float norm = dinv[r] * dinv[cdst];
        const float v    = h[(size_t)r * D + f] * norm;
        __hip_atomic_fetch_add(&out[(size_t)cdst * D + f], v,
                               __ATOMIC_RELAXED, __HIP_MEMORY_SCOPE_AGENT);
    }
}

// ---------------------------------------------------------------------------
extern "C" void kernel_launch(void* const* d_in, const int* in_sizes, int n_in,
                              void* d_out, int out_size, void* d_ws, size_t ws_size,
                              hipStream_t stream) {
    const float* x  = (const float*)d_in[0];   // [N, 64]
    const int*   ei = (const int*)d_in[1];     // [2, E] : row = ei[0:E], col = ei[E:2E]
    const float* W  = (const float*)d_in[2];   // [64, 64]
    const float* b  = (const float*)d_in[3];   // [64]
    float* out = (float*)d_out;                // [N, 64]

    // Workspace layout: h [N*64 floats] | deg/dinv [N floats]  (~26 MB)
    float* h   = (float*)d_ws;
    float* deg = h + (size_t)N_NODES * D;

    GCN_deg_init  <<<(N_NODES + 255) / 256, 256, 0, stream>>>(deg);
    GCN_deg_count <<<(N_EDGES + 255) / 256, 256, 0, stream>>>(ei + N_EDGES, deg);
    GCN_deg_to_dinv<<<(N_NODES + 255) / 256, 256, 0, stream>>>(deg);

    GCN_gemm_wmma <<<(N_NODES + 127) / 128, 256, 0, stream>>>(x, W, h);

    long total = (long)N_NODES * D;
    GCN_out_init  <<<(unsigned)((total + 255) / 256), 256, 0, stream>>>(h, deg, b, out);
    GCN_scatter   <<<(N_EDGES + 3) / 4, 256, 0, stream>>>(ei, h, deg, out);
}